// CompositeLoss_43791486550161
// MI455X (gfx1250) — compile-verified
//
#include <hip/hip_runtime.h>

typedef float v2f __attribute__((ext_vector_type(2)));
typedef float v8f __attribute__((ext_vector_type(8)));

#define NACC 13
static constexpr int Bn = 2, Cn = 3, Dn = 128, Hn = 128, Wn = 128;
static constexpr int NBLK  = Bn * Dn * Hn;   // 32768 phase-1 blocks (one per (b,d,h) row)
static constexpr int NMID  = 64;             // phase-2 blocks
static constexpr int CHUNK = NBLK / NMID;    // 512 partials per phase-2 block
static constexpr float EPSF = 1e-8f;

// Full-wave (32 lane) sum using V_WMMA_F32_16X16X4_F32.
// A[m][0] = x(lane m), A[m][2] = x(lane m+16) (ISA 16x4 f32 A layout),
// A[m][1]=A[m][3]=0; B = ones (layout independent) -> D[m][n] = x(m)+x(m+16).
// Summing D's 8 VGPRs per lane gives lanes 0-15: sum(m=0..7), lanes 16-31:
// sum(m=8..15); one shfl_xor(16) leaves the full 32-lane sum in EVERY lane.
// Requires EXEC == all ones (call only from non-divergent code).
__device__ __forceinline__ float wave_sum32(float x) {
    v2f a; a[0] = x;   a[1] = 0.0f;
    v2f b; b[0] = 1.0f; b[1] = 1.0f;
    v8f c = {0.f, 0.f, 0.f, 0.f, 0.f, 0.f, 0.f, 0.f};
    v8f d = __builtin_amdgcn_wmma_f32_16x16x4_f32(
        /*neg_a=*/false, a, /*neg_b=*/false, b,
        /*c_mod=*/(short)0, c, /*reuse_a=*/false, /*reuse_b=*/false);
    float s = ((d[0] + d[1]) + (d[2] + d[3])) + ((d[4] + d[5]) + (d[6] + d[7]));
    s += __shfl_xor(s, 16, 32);
    return s;
}

// Phase 1: one block per (b,d,h) row, lane = w. 13 partial sums per block.
__global__ __launch_bounds__(128) void loss_partial_kernel(
    const float* __restrict__ pred, const float* __restrict__ target,
    const float* __restrict__ mask, float* __restrict__ partial) {
    const int tid = threadIdx.x;               // w in [0,128)
    const int s   = blockIdx.x;                // b*Dn*Hn + d*Hn + h
    const int h   = s % Hn;
    const int d   = (s / Hn) % Dn;
    const int b   = s / (Hn * Dn);
    const int w   = tid;

    const int  mbase = ((b * Dn + d) * Hn + h) * Wn + w;
    const bool hasW  = (w < Wn - 1);
    const bool hasH  = (h < Hn - 1);
    const bool hasD  = (d < Dn - 1);

    const float m   = mask[mbase];
    const float mw1 = hasW ? mask[mbase + 1]       : 0.0f;
    const float mh1 = hasH ? mask[mbase + Wn]      : 0.0f;
    const float md1 = hasD ? mask[mbase + Wn * Hn] : 0.0f;
    const float inv = 1.0f - m;
    const float mx  = fminf(m, md1);   // 0 at boundary (md1=0)
    const float my  = fminf(m, mh1);
    const float mz  = fminf(m, mw1);

    float a_mae = 0.f, a_mse = 0.f, a_bg = 0.f;
    float a_gx = 0.f, a_gy = 0.f, a_gz = 0.f;
    float a_tx = 0.f, a_ty = 0.f, a_tz = 0.f;

#pragma unroll
    for (int ch = 0; ch < Cn; ++ch) {
        const int base = (((b * Cn + ch) * Dn + d) * Hn + h) * Wn + w;
        const float p    = pred[base];
        const float t    = target[base];
        const float pin  = p * m;
        const float tin  = t * m;
        const float diff = pin - tin;
        a_mae += fabsf(diff) * m;
        a_mse += diff * diff * m;
        a_bg  += fabsf(p) * inv;
        if (hasD) {
            const float p1 = pred[base + Wn * Hn] * md1;
            const float t1 = target[base + Wn * Hn] * md1;
            const float gp = p1 - pin;
            const float gt = t1 - tin;
            a_gx += fabsf(gp - gt) * mx;
            a_tx += fabsf(gp) * mx;
        }
        if (hasH) {
            const float p1 = pred[base + Wn] * mh1;
            const float t1 = target[base + Wn] * mh1;
            const float gp = p1 - pin;
            const float gt = t1 - tin;
            a_gy += fabsf(gp - gt) * my;
            a_ty += fabsf(gp) * my;
        }
        if (hasW) {
            const float p1 = pred[base + 1] * mw1;
            const float t1 = target[base + 1] * mw1;
            const float gp = p1 - pin;
            const float gt = t1 - tin;
            a_gz += fabsf(gp - gt) * mz;
            a_tz += fabsf(gp) * mz;
        }
    }

    float acc[NACC] = {a_mae, a_mse, a_bg,
                       a_gx,  a_gy,  a_gz,
                       a_tx,  a_ty,  a_tz,
                       m,     mx,    my,   mz};

    __shared__ float lds[4 * NACC];
    const int wave = tid >> 5;
    const int lane = tid & 31;
#pragma unroll
    for (int j = 0; j < NACC; ++j) {
        const float ws = wave_sum32(acc[j]);   // full EXEC here
        if (lane == 0) lds[wave * NACC + j] = ws;
    }
    __syncthreads();
    if (tid < NACC) {
        const float v = lds[tid] + lds[NACC + tid] + lds[2 * NACC + tid] + lds[3 * NACC + tid];
        partial[tid * NBLK + blockIdx.x] = v;   // SoA: coalesced phase-2 reads
    }
}

// Phase 2: 64 blocks fold 512 first-level partials each (parallel tail,
// avoids a serial single-WGP 1.7 MB read). Writes 13 x 64 second-level sums.
__global__ __launch_bounds__(256) void loss_mid_kernel(
    const float* __restrict__ partial, float* __restrict__ p2) {
    const int tid  = threadIdx.x;
    const int bk   = blockIdx.x;
    const int base = bk * CHUNK;               // CHUNK = 512 = 2 * 256
    float acc[NACC];
#pragma unroll
    for (int j = 0; j < NACC; ++j) {
        acc[j] = partial[j * NBLK + base + tid] +
                 partial[j * NBLK + base + 256 + tid];
    }

    __shared__ float lds[8 * NACC];
    const int wave = tid >> 5;
    const int lane = tid & 31;
#pragma unroll
    for (int j = 0; j < NACC; ++j) {
        const float ws = wave_sum32(acc[j]);   // full EXEC here
        if (lane == 0) lds[wave * NACC + j] = ws;
    }
    __syncthreads();
    if (tid < NACC) {
        float v = 0.0f;
#pragma unroll
        for (int wv = 0; wv < 8; ++wv) v += lds[wv * NACC + tid];
        p2[tid * NMID + bk] = v;
    }
}

// Phase 3: one wave folds the 13 x 64 second-level sums (wave_sum32 leaves
// the full sum in every lane -> lane 0 evaluates the weighted loss; no LDS,
// no barriers).
__global__ __launch_bounds__(32) void loss_final_kernel(
    const float* __restrict__ p2, float* __restrict__ out) {
    const int lane = threadIdx.x;
    float tot[NACC];
#pragma unroll
    for (int j = 0; j < NACC; ++j) {
        const float a = p2[j * NMID + lane] + p2[j * NMID + lane + 32];
        tot[j] = wave_sum32(a);                // full EXEC here
    }
    if (lane == 0) {
        const float s_mae = tot[0], s_mse = tot[1], s_bg = tot[2];
        const float s_gx = tot[3], s_gy = tot[4], s_gz = tot[5];
        const float s_tx = tot[6], s_ty = tot[7], s_tz = tot[8];
        const float s_m  = tot[9], s_mx = tot[10], s_my = tot[11], s_mz = tot[12];
        const float Nmask   = (float)(Bn * Dn * Hn * Wn);
        const float inv_sum = Nmask - s_m;
        const float dm = s_m  * (float)Cn + EPSF;
        const float dx = s_mx * (float)Cn + EPSF;
        const float dy = s_my * (float)Cn + EPSF;
        const float dz = s_mz * (float)Cn + EPSF;
        const float l_mae  = s_mae / dm;
        const float l_grad = s_gx / dx + s_gy / dy + s_gz / dz;
        const float l_tv   = s_tx / dx + s_ty / dy + s_tz / dz;
        const float l_bg   = s_bg / (inv_sum * (float)Cn + EPSF);
        const float l_mse  = s_mse / dm;
        out[0] = 1.0f * l_mae + 0.1f * l_grad + 0.002f * l_tv + 0.15f * l_bg + 1.0f * l_mse;
    }
}

extern "C" void kernel_launch(void* const* d_in, const int* in_sizes, int n_in,
                              void* d_out, int out_size, void* d_ws, size_t ws_size,
                              hipStream_t stream) {
    (void)in_sizes; (void)n_in; (void)out_size; (void)ws_size;
    const float* pred   = (const float*)d_in[0];
    const float* target = (const float*)d_in[1];
    const float* mask   = (const float*)d_in[2];
    float* partial = (float*)d_ws;                       // 13 * 32768 floats (1.7 MB)
    float* p2      = partial + (size_t)NACC * NBLK;      // 13 * 64 floats

    loss_partial_kernel<<<dim3(NBLK), dim3(128), 0, stream>>>(pred, target, mask, partial);
    loss_mid_kernel<<<dim3(NMID), dim3(256), 0, stream>>>(partial, p2);
    loss_final_kernel<<<dim3(1), dim3(32), 0, stream>>>(p2, (float*)d_out);
}